// UltraEfficientRouter_10909216932607
// MI455X (gfx1250) — compile-verified
//
#include <hip/hip_runtime.h>
#include <hip/hip_bf16.h>
#include <math.h>

// Problem constants (from reference)
#define B_   16
#define C_   256
#define HW_  224
#define POOL_ 8
#define PH_  28          // pooled H/W
#define NPIX_ 784        // 28*28
#define RC_  16
#define E_   8
#define GN1G_ 8          // groups (32 channels each)
#define GN2G_ 4          // groups (4 channels each)

typedef float v2f __attribute__((ext_vector_type(2)));
typedef float v8f __attribute__((ext_vector_type(8)));

// silu via hardware v_rcp_f32 (TRANS op, ~1 ulp): avoids the IEEE div_fixup chain
__device__ __forceinline__ float silu_fast(float t) {
    return t * __builtin_amdgcn_rcpf(1.0f + __expf(-t));
}

// ---------------------------------------------------------------------------
// Kernel 0: zero the atomic GN statistics buffers (must be done every launch)
// ---------------------------------------------------------------------------
__global__ void zero_stats_kernel(float* __restrict__ gn1s, float* __restrict__ gn2s) {
    int t = threadIdx.x;
    if (t < B_ * GN1G_ * 2) gn1s[t] = 0.0f;   // 256
    if (t < B_ * GN2G_ * 2) gn2s[t] = 0.0f;   // 128
}

// ---------------------------------------------------------------------------
// Kernel 1: fused 8x8 avg-pool + depthwise 3x3 conv + GN1 partial statistics
// One block per (b, c) plane. Pooled plane lives only in LDS (30x30 padded).
// This kernel is the bandwidth roofline: 822 MB read of x, 13 MB write of h.
// ---------------------------------------------------------------------------
__global__ void pool_conv_gn1_kernel(const float* __restrict__ x,
                                     const float* __restrict__ w_dw,
                                     float* __restrict__ h,
                                     float* __restrict__ gn1_stats) {
    const int plane = blockIdx.x;        // b*C + c
    const int c = plane & (C_ - 1);
    const int b = plane >> 8;
    const int tid = threadIdx.x;

    __shared__ float sp[30 * 30];        // pooled plane with 1-wide zero halo
    __shared__ float rs[256];
    __shared__ float rq[256];

    for (int i = tid; i < 30 * 30; i += 256) sp[i] = 0.0f;
    __syncthreads();

    const float* xp = x + (size_t)plane * (HW_ * HW_);

    // ---- pool: each thread computes pooled pixels p = tid, tid+256, ...
    for (int p = tid; p < NPIX_; p += 256) {
        const int oh = p / PH_;
        const int ow = p - oh * PH_;
        const float* base = xp + (oh * POOL_) * HW_ + ow * POOL_;
        float s = 0.0f;
#pragma unroll
        for (int r = 0; r < POOL_; ++r) {
            const float4* row = (const float4*)(base + r * HW_);
            float4 a = row[0];
            float4 q = row[1];
            s += a.x + a.y + a.z + a.w + q.x + q.y + q.z + q.w;
        }
        sp[(oh + 1) * 30 + (ow + 1)] = s * (1.0f / 64.0f);
    }
    __syncthreads();

    // ---- depthwise 3x3, padding=1, from LDS
    float w[9];
#pragma unroll
    for (int i = 0; i < 9; ++i) w[i] = w_dw[c * 9 + i];

    float lsum = 0.0f, lsq = 0.0f;
    float* hp = h + (size_t)plane * NPIX_;
    for (int p = tid; p < NPIX_; p += 256) {
        const int oh = p / PH_;
        const int ow = p - oh * PH_;
        const float* s0 = &sp[oh * 30 + ow];   // top-left of 3x3 window
        float acc = s0[0]  * w[0] + s0[1]  * w[1] + s0[2]  * w[2]
                  + s0[30] * w[3] + s0[31] * w[4] + s0[32] * w[5]
                  + s0[60] * w[6] + s0[61] * w[7] + s0[62] * w[8];
        hp[p] = acc;
        lsum += acc;
        lsq  += acc * acc;
    }

    // ---- block reduce sum / sumsq, accumulate GN1 stats
    rs[tid] = lsum;
    rq[tid] = lsq;
    __syncthreads();
    for (int off = 128; off > 0; off >>= 1) {
        if (tid < off) { rs[tid] += rs[tid + off]; rq[tid] += rq[tid + off]; }
        __syncthreads();
    }
    if (tid == 0) {
        const int g = c >> 5;                    // 32 channels per group
        atomicAdd(&gn1_stats[(b * GN1G_ + g) * 2 + 0], rs[0]);
        atomicAdd(&gn1_stats[(b * GN1G_ + g) * 2 + 1], rq[0]);
    }
}

// ---------------------------------------------------------------------------
// Kernel 2: GN1 + silu + pointwise conv C(256) -> RC(16) via V_WMMA_F32_16X16X4_F32
// One wave (32 threads) per 16x16 output tile: M = 16 output channels,
// N = 16 pixels, K loop over 256 input channels in steps of 4 (unroll x2).
// GN1 is folded to h*scale[c] + shift[c], precomputed in LDS so the K-loop
// only issues the two real data loads per lane per step.
// ---------------------------------------------------------------------------
__global__ void pw1_wmma_gn2_kernel(const float* __restrict__ h,
                                    const float* __restrict__ w_pw1,
                                    const float* __restrict__ gn1_g,
                                    const float* __restrict__ gn1_b,
                                    const float* __restrict__ gn1_stats,
                                    float* __restrict__ out2,
                                    float* __restrict__ gn2_stats) {
    const int blk  = blockIdx.x;          // b*49 + tile
    const int b    = blk / 49;
    const int tile = blk - b * 49;
    const int lane = threadIdx.x;         // 0..31, wave32
    const int n     = lane & 15;          // pixel within tile / row M for A
    const int khalf = (lane >> 4) * 2;    // K offset 0 or 2 within the 4-wide step
    const int p     = tile * 16 + n;      // pixel index 0..783

    __shared__ float mean1[GN1G_];
    __shared__ float rstd1[GN1G_];
    __shared__ float sc[C_];              // per-channel GN1 scale = rstd*gamma
    __shared__ float sh[C_];              // per-channel GN1 shift = beta - mean*scale

    if (lane < GN1G_) {
        const float invN = 1.0f / (32.0f * (float)NPIX_);
        float s = gn1_stats[(b * GN1G_ + lane) * 2 + 0];
        float q = gn1_stats[(b * GN1G_ + lane) * 2 + 1];
        float m = s * invN;
        float v = q * invN - m * m;
        mean1[lane] = m;
        rstd1[lane] = rsqrtf(v + 1e-5f);
    }
    __syncthreads();
    for (int c = lane; c < C_; c += 32) {
        const int g = c >> 5;
        float s = rstd1[g] * gn1_g[c];
        sc[c] = s;
        sh[c] = gn1_b[c] - mean1[g] * s;
    }
    __syncthreads();

    const float* hb = h + (size_t)b * C_ * NPIX_;
    v8f acc = {0.f, 0.f, 0.f, 0.f, 0.f, 0.f, 0.f, 0.f};

#pragma unroll 2
    for (int kk = 0; kk < C_; kk += 4) {
        // A fragment: w_pw1[m][k], m = lane&15; lane<16 holds K={kk,kk+1},
        // lane>=16 holds K={kk+2,kk+3}  (ISA 32-bit 16x4 A layout)
        const int c0 = kk + khalf;
        const int c1 = c0 + 1;
        v2f a;
        a.x = w_pw1[n * C_ + c0];
        a.y = w_pw1[n * C_ + c1];

        // B fragment: silu(GN1(h))[k][p]; mirrored layout (lane = column N)
        float v0 = hb[(size_t)c0 * NPIX_ + p] * sc[c0] + sh[c0];
        float v1 = hb[(size_t)c1 * NPIX_ + p] * sc[c1] + sh[c1];
        v2f bf;
        bf.x = silu_fast(v0);
        bf.y = silu_fast(v1);

        acc = __builtin_amdgcn_wmma_f32_16x16x4_f32(false, a, false, bf,
                                                    (short)0, acc, false, false);
    }

    // ---- store D tile + GN2 partial stats (constant-index epilogue: the
    // lane<16 / lane>=16 halves are written with compile-time M so no
    // register-select chains are generated)
    float* ob = out2 + (size_t)b * RC_ * NPIX_;
    float gs[GN2G_] = {0.f, 0.f, 0.f, 0.f};
    float gq[GN2G_] = {0.f, 0.f, 0.f, 0.f};
    if (lane < 16) {
#pragma unroll
        for (int i = 0; i < 8; ++i) {              // M = i (0..7) -> groups 0,1
            float v = acc[i];
            ob[(size_t)i * NPIX_ + p] = v;
            gs[i >> 2] += v;
            gq[i >> 2] += v * v;
        }
    } else {
#pragma unroll
        for (int i = 0; i < 8; ++i) {              // M = i+8 (8..15) -> groups 2,3
            float v = acc[i];
            ob[(size_t)(i + 8) * NPIX_ + p] = v;
            gs[(i + 8) >> 2] += v;
            gq[(i + 8) >> 2] += v * v;
        }
    }
#pragma unroll
    for (int g = 0; g < GN2G_; ++g) {
        float s = gs[g], q = gq[g];
        for (int off = 16; off > 0; off >>= 1) {
            s += __shfl_down(s, off, 32);
            q += __shfl_down(q, off, 32);
        }
        if (lane == 0) {
            atomicAdd(&gn2_stats[(b * GN2G_ + g) * 2 + 0], s);
            atomicAdd(&gn2_stats[(b * GN2G_ + g) * 2 + 1], q);
        }
    }
}

// ---------------------------------------------------------------------------
// Kernel 3: GN2 + silu + pointwise 16->8 (+bias) + softmax + spatial mean +
// top-2 + renormalize. One block per batch element.
// ---------------------------------------------------------------------------
__global__ void finalize_kernel(const float* __restrict__ out2,
                                const float* __restrict__ gn2_stats,
                                const float* __restrict__ gn2_g,
                                const float* __restrict__ gn2_b,
                                const float* __restrict__ w_pw2,
                                const float* __restrict__ b_pw2,
                                float* __restrict__ outv,
                                int* __restrict__ outi) {
    const int b = blockIdx.x;
    const int tid = threadIdx.x;

    __shared__ float sc2[RC_], sh2[RC_];
    __shared__ float wsm[E_ * RC_], bsm[E_];
    __shared__ float pooled[E_];

    if (tid < GN2G_) {
        const float invN = 1.0f / (4.0f * (float)NPIX_);
        float s = gn2_stats[(b * GN2G_ + tid) * 2 + 0];
        float q = gn2_stats[(b * GN2G_ + tid) * 2 + 1];
        float m = s * invN;
        float v = q * invN - m * m;
        float r = rsqrtf(v + 1e-5f);
        // fold GN2 affine per channel for this group
#pragma unroll
        for (int j = 0; j < 4; ++j) {
            int ch = tid * 4 + j;
            float scale = r * gn2_g[ch];
            sc2[ch] = scale;
            sh2[ch] = gn2_b[ch] - m * scale;
        }
    }
    if (tid < E_ * RC_) wsm[tid] = w_pw2[tid];
    if (tid < E_) { bsm[tid] = b_pw2[tid]; pooled[tid] = 0.0f; }
    __syncthreads();

    const float* ob = out2 + (size_t)b * RC_ * NPIX_;
    float lp[E_] = {0.f, 0.f, 0.f, 0.f, 0.f, 0.f, 0.f, 0.f};

    for (int p = tid; p < NPIX_; p += 256) {
        float v[RC_];
#pragma unroll
        for (int r = 0; r < RC_; ++r) {
            float t = ob[r * NPIX_ + p] * sc2[r] + sh2[r];
            v[r] = silu_fast(t);
        }
        float l[E_];
        float mx = -1e30f;
#pragma unroll
        for (int e = 0; e < E_; ++e) {
            float a = bsm[e];
#pragma unroll
            for (int r = 0; r < RC_; ++r) a += v[r] * wsm[e * RC_ + r];
            l[e] = a;
            mx = fmaxf(mx, a);
        }
        float ssum = 0.0f;
#pragma unroll
        for (int e = 0; e < E_; ++e) { l[e] = __expf(l[e] - mx); ssum += l[e]; }
        float inv = __builtin_amdgcn_rcpf(ssum);
#pragma unroll
        for (int e = 0; e < E_; ++e) lp[e] += l[e] * inv;
    }
#pragma unroll
    for (int e = 0; e < E_; ++e) atomicAdd(&pooled[e], lp[e]);
    __syncthreads();

    if (tid == 0) {
        float pv[E_];
#pragma unroll
        for (int e = 0; e < E_; ++e) pv[e] = pooled[e] * (1.0f / (float)NPIX_);
        // top-2, ties -> lowest index (matches jax.lax.top_k)
        int i0 = 0;
        for (int e = 1; e < E_; ++e) if (pv[e] > pv[i0]) i0 = e;
        int i1 = -1;
        for (int e = 0; e < E_; ++e) {
            if (e == i0) continue;
            if (i1 < 0 || pv[e] > pv[i1]) i1 = e;
        }
        float s = pv[i0] + pv[i1] + 1e-9f;   // exact division for final outputs
        outv[b * 2 + 0] = pv[i0] / s;
        outv[b * 2 + 1] = pv[i1] / s;
        outi[b * 2 + 0] = i0;
        outi[b * 2 + 1] = i1;
    }
}

// ---------------------------------------------------------------------------
extern "C" void kernel_launch(void* const* d_in, const int* in_sizes, int n_in,
                              void* d_out, int out_size, void* d_ws, size_t ws_size,
                              hipStream_t stream) {
    const float* x     = (const float*)d_in[0];
    const float* w_dw  = (const float*)d_in[1];
    const float* gn1_g = (const float*)d_in[2];
    const float* gn1_b = (const float*)d_in[3];
    const float* w_pw1 = (const float*)d_in[4];
    const float* gn2_g = (const float*)d_in[5];
    const float* gn2_b = (const float*)d_in[6];
    const float* w_pw2 = (const float*)d_in[7];
    const float* b_pw2 = (const float*)d_in[8];

    // workspace layout (floats): h | out2 | gn1_stats | gn2_stats  (~13.7 MB)
    float* ws    = (float*)d_ws;
    float* h     = ws;                                  // B*C*784
    float* out2  = h + (size_t)B_ * C_ * NPIX_;         // B*RC*784
    float* gn1s  = out2 + (size_t)B_ * RC_ * NPIX_;     // B*8*2
    float* gn2s  = gn1s + B_ * GN1G_ * 2;               // B*4*2

    float* outv = (float*)d_out;            // topk_vals: (16,2,1,1) fp32
    int*   outi = (int*)d_out + B_ * 2;     // topk_idx : (16,2,1,1) int32

    zero_stats_kernel<<<1, 256, 0, stream>>>(gn1s, gn2s);
    pool_conv_gn1_kernel<<<B_ * C_, 256, 0, stream>>>(x, w_dw, h, gn1s);
    pw1_wmma_gn2_kernel<<<B_ * 49, 32, 0, stream>>>(h, w_pw1, gn1_g, gn1_b,
                                                    gn1s, out2, gn2s);
    finalize_kernel<<<B_, 256, 0, stream>>>(out2, gn2s, gn2_g, gn2_b,
                                            w_pw2, b_pw2, outv, outi);
}